// InvResMLP_56624848831028
// MI455X (gfx1250) — compile-verified
//
#include <hip/hip_runtime.h>
#include <hip/hip_bf16.h>
#include <math.h>

// ---------------- problem constants (from reference) ----------------
#define BATCH 2
#define CDIM  64
#define NPTS  4096
#define MDIM  256      // 4*C
#define NS    32       // NSAMPLE
#define R2    0.04f    // RADIUS^2
#define NGRP  4
#define EPS_GN 1e-5f

typedef float v2f __attribute__((ext_vector_type(2)));
typedef float v8f __attribute__((ext_vector_type(8)));

__device__ __forceinline__ float gelu_exact(float v) {
    return 0.5f * v * (1.0f + erff(v * 0.70710678118654752f));
}

__device__ __forceinline__ float wave_sum32(float v) {
    #pragma unroll
    for (int off = 16; off >= 1; off >>= 1) v += __shfl_xor(v, off, 32);
    return v;
}

// ---------------- K1: ball query ----------------
// grid (NPTS/256, BATCH), block 256. Each thread owns one query point n.
__global__ void k_ball_query(const float* __restrict__ pos, int* __restrict__ idx) {
    const int b   = blockIdx.y;
    const int tid = threadIdx.x;
    const int n   = blockIdx.x * 256 + tid;
    const float* px = pos + (size_t)b * 3 * NPTS;
    const float* py = px + NPTS;
    const float* pz = px + 2 * NPTS;

    const float qx = px[n], qy = py[n], qz = pz[n];
    int* row = idx + ((size_t)b * NPTS + n) * NS;

    __shared__ float sx[256], sy[256], sz[256];
    int cnt = 0, first = n;   // self-distance is 0, so n always qualifies

    for (int t0 = 0; t0 < NPTS; t0 += 256) {
        sx[tid] = px[t0 + tid];
        sy[tid] = py[t0 + tid];
        sz[tid] = pz[t0 + tid];
        __syncthreads();
        if (cnt < NS) {
            for (int j = 0; j < 256; ++j) {
                float dx = qx - sx[j], dy = qy - sy[j], dz = qz - sz[j];
                float d2 = dx * dx + dy * dy + dz * dz;
                if (d2 <= R2) {
                    if (cnt == 0) first = t0 + j;
                    row[cnt] = t0 + j;
                    if (++cnt == NS) break;
                }
            }
        }
        __syncthreads();
    }
    for (int k = cnt; k < NS; ++k) row[k] = first;
}

// ---------------- K2: GEMM1 h = w1*x + b1  (WMMA f32 16x16x4) ----------------
// grid (NPTS/16, MDIM/16, BATCH), block 32 (one wave = one 16x16 tile).
__global__ void k_gemm1(const float* __restrict__ w1, const float* __restrict__ b1,
                        const float* __restrict__ x, float* __restrict__ h,
                        float* __restrict__ stats1) {
    const int lane = threadIdx.x;
    const int ln   = lane & 15;
    const int hi   = (lane >> 4) & 1;          // lanes 16..31
    const int n0   = blockIdx.x * 16;
    const int m0   = blockIdx.y * 16;
    const int b    = blockIdx.z;
    const float* xb = x + (size_t)b * CDIM * NPTS;

    v8f c = {};
    #pragma unroll
    for (int k = 0; k < CDIM; k += 4) {
        const int ka = k + (hi ? 2 : 0);
        v2f a, bb;
        a.x  = w1[(m0 + ln) * CDIM + ka];
        a.y  = w1[(m0 + ln) * CDIM + ka + 1];
        bb.x = xb[(size_t)ka * NPTS + n0 + ln];
        bb.y = xb[(size_t)(ka + 1) * NPTS + n0 + ln];
        c = __builtin_amdgcn_wmma_f32_16x16x4_f32(false, a, false, bb,
                                                  (short)0, c, false, false);
    }

    float gs = 0.f, gq = 0.f;
    #pragma unroll
    for (int r = 0; r < 8; ++r) {
        const int m = m0 + r + (hi ? 8 : 0);
        float v = c[r] + b1[m];
        h[((size_t)b * MDIM + m) * NPTS + n0 + ln] = v;
        gs += v; gq += v * v;
    }
    gs = wave_sum32(gs);
    gq = wave_sum32(gq);
    if (lane == 0) {
        const int g = m0 >> 6;                 // 64 channels per group
        atomicAdd(&stats1[(b * NGRP + g) * 2 + 0], gs);
        atomicAdd(&stats1[(b * NGRP + g) * 2 + 1], gq);
    }
}

// ---------------- K3: GN1 + GELU, write feats in (B,N,M) ----------------
__global__ void k_gn1_gelu(const float* __restrict__ h, const float* __restrict__ g1,
                           const float* __restrict__ be1, const float* __restrict__ stats1,
                           float* __restrict__ feats) {
    const size_t t = (size_t)blockIdx.x * 256 + threadIdx.x;   // over B*M*N, n fastest
    const int n = (int)(t % NPTS);
    const int m = (int)((t / NPTS) % MDIM);
    const int b = (int)(t / ((size_t)NPTS * MDIM));
    const int g = m >> 6;
    const float cnt  = (float)((MDIM / NGRP) * NPTS);
    const float mean = stats1[(b * NGRP + g) * 2 + 0] / cnt;
    const float var  = stats1[(b * NGRP + g) * 2 + 1] / cnt - mean * mean;
    const float rstd = rsqrtf(var + EPS_GN);
    float v = (h[t] - mean) * rstd * g1[m] + be1[m];
    feats[((size_t)b * NPTS + n) * MDIM + m] = gelu_exact(v);
}

// ---------------- K4: gather + wd scale, per-channel max/min + GN2 stats ----------------
// grid (NPTS, BATCH), block 256 (thread = channel c). feats rows are contiguous.
__global__ void k_gather_minmax(const float* __restrict__ feats, const int* __restrict__ idx,
                                const float* __restrict__ wd,
                                float* __restrict__ gmax, float* __restrict__ gmin,
                                float* __restrict__ stats2) {
    const int c = threadIdx.x;
    const int n = blockIdx.x;
    const int b = blockIdx.y;
    __shared__ int sj[NS];
    if (c < NS) sj[c] = idx[((size_t)b * NPTS + n) * NS + c];
    __syncthreads();

    const float w = wd[c];
    const float* fb = feats + (size_t)b * NPTS * MDIM;
    float mx = -INFINITY, mn = INFINITY, s = 0.f, q = 0.f;
    #pragma unroll 4
    for (int k = 0; k < NS; ++k) {
        if (k + 1 < NS)
            __builtin_prefetch(&fb[(size_t)sj[k + 1] * MDIM + c], 0, 1); // global_prefetch_b8
        float v = fb[(size_t)sj[k] * MDIM + c] * w;
        mx = fmaxf(mx, v); mn = fminf(mn, v);
        s += v; q += v * v;
    }
    gmax[((size_t)b * NPTS + n) * MDIM + c] = mx;
    gmin[((size_t)b * NPTS + n) * MDIM + c] = mn;

    // segmented reduction: 4 groups of 64 contiguous channels
    __shared__ float rs[256], rq[256];
    rs[c] = s; rq[c] = q;
    __syncthreads();
    #pragma unroll
    for (int off = 32; off >= 1; off >>= 1) {
        if ((c & 63) < off) { rs[c] += rs[c + off]; rq[c] += rq[c + off]; }
        __syncthreads();
    }
    if ((c & 63) == 0) {
        const int g = c >> 6;
        atomicAdd(&stats2[(b * NGRP + g) * 2 + 0], rs[c]);
        atomicAdd(&stats2[(b * NGRP + g) * 2 + 1], rq[c]);
    }
}

// ---------------- K5: GN2 + max-over-samples + GELU -> h2t (B,N,M) ----------------
__global__ void k_gn2_max_gelu(const float* __restrict__ gmax, const float* __restrict__ gmin,
                               const float* __restrict__ gd, const float* __restrict__ bed,
                               const float* __restrict__ stats2, float* __restrict__ h2t) {
    const int m = threadIdx.x;
    const int n = blockIdx.x;
    const int b = blockIdx.y;
    const int g = m >> 6;
    const float cnt  = (float)((MDIM / NGRP) * NPTS) * (float)NS;
    const float mean = stats2[(b * NGRP + g) * 2 + 0] / cnt;
    const float var  = stats2[(b * NGRP + g) * 2 + 1] / cnt - mean * mean;
    const float rstd = rsqrtf(var + EPS_GN);
    const float sc = gd[m];
    const size_t o = ((size_t)b * NPTS + n) * MDIM + m;
    // max(GN(v)) over k: affine with slope rstd*sc -> pick max if sc>=0 else min
    float v = (sc >= 0.f) ? gmax[o] : gmin[o];
    v = (v - mean) * rstd * sc + bed[m];
    h2t[o] = gelu_exact(v);
}

// ---------------- K6: GEMM2 out = w2*h2 + b2 (WMMA f32), GN3 stats ----------------
// grid (NPTS/16, CDIM/16, BATCH), block 32. h2 stored transposed as (B,N,M).
__global__ void k_gemm2(const float* __restrict__ w2, const float* __restrict__ b2,
                        const float* __restrict__ h2t, float* __restrict__ outp,
                        float* __restrict__ stats3) {
    const int lane = threadIdx.x;
    const int ln   = lane & 15;
    const int hi   = (lane >> 4) & 1;
    const int n0   = blockIdx.x * 16;
    const int c0   = blockIdx.y * 16;
    const int b    = blockIdx.z;
    const float* hb = h2t + (size_t)b * NPTS * MDIM;

    v8f c = {};
    for (int k = 0; k < MDIM; k += 4) {
        const int ka = k + (hi ? 2 : 0);
        v2f a, bb;
        a.x  = w2[(c0 + ln) * MDIM + ka];
        a.y  = w2[(c0 + ln) * MDIM + ka + 1];
        bb.x = hb[(size_t)(n0 + ln) * MDIM + ka];      // h2[k][n] = h2t[n][k]
        bb.y = hb[(size_t)(n0 + ln) * MDIM + ka + 1];
        c = __builtin_amdgcn_wmma_f32_16x16x4_f32(false, a, false, bb,
                                                  (short)0, c, false, false);
    }

    float gs = 0.f, gq = 0.f;
    #pragma unroll
    for (int r = 0; r < 8; ++r) {
        const int cc = c0 + r + (hi ? 8 : 0);
        float v = c[r] + b2[cc];
        outp[((size_t)b * CDIM + cc) * NPTS + n0 + ln] = v;
        gs += v; gq += v * v;
    }
    gs = wave_sum32(gs);
    gq = wave_sum32(gq);
    if (lane == 0) {
        const int g = c0 >> 4;                 // 16 channels per group
        atomicAdd(&stats3[(b * NGRP + g) * 2 + 0], gs);
        atomicAdd(&stats3[(b * NGRP + g) * 2 + 1], gq);
    }
}

// ---------------- K7: GN3 + residual ----------------
__global__ void k_gn3_residual(const float* __restrict__ outp, const float* __restrict__ g2,
                               const float* __restrict__ be2, const float* __restrict__ stats3,
                               const float* __restrict__ x, float* __restrict__ out) {
    const size_t t = (size_t)blockIdx.x * 256 + threadIdx.x;   // over B*C*N, n fastest
    const int cc = (int)((t / NPTS) % CDIM);
    const int b  = (int)(t / ((size_t)NPTS * CDIM));
    const int g  = cc >> 4;
    const float cnt  = (float)((CDIM / NGRP) * NPTS);
    const float mean = stats3[(b * NGRP + g) * 2 + 0] / cnt;
    const float var  = stats3[(b * NGRP + g) * 2 + 1] / cnt - mean * mean;
    const float rstd = rsqrtf(var + EPS_GN);
    out[t] = (outp[t] - mean) * rstd * g2[cc] + be2[cc] + x[t];
}

// ---------------- host ----------------
extern "C" void kernel_launch(void* const* d_in, const int* in_sizes, int n_in,
                              void* d_out, int out_size, void* d_ws, size_t ws_size,
                              hipStream_t stream) {
    (void)in_sizes; (void)n_in; (void)out_size; (void)ws_size;
    const float* x   = (const float*)d_in[0];
    const float* pos = (const float*)d_in[1];
    const float* w1  = (const float*)d_in[2];
    const float* b1  = (const float*)d_in[3];
    const float* g1  = (const float*)d_in[4];
    const float* be1 = (const float*)d_in[5];
    const float* wd  = (const float*)d_in[6];
    const float* gd  = (const float*)d_in[7];
    const float* bed = (const float*)d_in[8];
    const float* w2  = (const float*)d_in[9];
    const float* b2  = (const float*)d_in[10];
    const float* g2  = (const float*)d_in[11];
    const float* be2 = (const float*)d_in[12];

    char* ws = (char*)d_ws;
    const size_t SZ_IDX   = (size_t)BATCH * NPTS * NS * sizeof(int);     // 1 MB
    const size_t SZ_BMN   = (size_t)BATCH * MDIM * NPTS * sizeof(float); // 8 MB
    const size_t SZ_BCN   = (size_t)BATCH * CDIM * NPTS * sizeof(float); // 2 MB
    int*   idx   = (int*)ws;                       size_t o = SZ_IDX;
    float* h     = (float*)(ws + o);               o += SZ_BMN;
    float* feats = (float*)(ws + o);               o += SZ_BMN;
    float* gmax  = (float*)(ws + o);               o += SZ_BMN;
    float* gmin  = (float*)(ws + o);               o += SZ_BMN;
    float* h2t   = (float*)(ws + o);               o += SZ_BMN;
    float* outp  = (float*)(ws + o);               o += SZ_BCN;
    float* stats = (float*)(ws + o);               // 3 * 16 floats
    float* stats1 = stats, *stats2 = stats + 16, *stats3 = stats + 32;

    hipMemsetAsync(stats, 0, 64 * sizeof(float), stream);

    k_ball_query   <<<dim3(NPTS / 256, BATCH), 256, 0, stream>>>(pos, idx);
    k_gemm1        <<<dim3(NPTS / 16, MDIM / 16, BATCH), 32, 0, stream>>>(w1, b1, x, h, stats1);
    k_gn1_gelu     <<<(BATCH * MDIM * NPTS) / 256, 256, 0, stream>>>(h, g1, be1, stats1, feats);
    k_gather_minmax<<<dim3(NPTS, BATCH), 256, 0, stream>>>(feats, idx, wd, gmax, gmin, stats2);
    k_gn2_max_gelu <<<dim3(NPTS, BATCH), 256, 0, stream>>>(gmax, gmin, gd, bed, stats2, h2t);
    k_gemm2        <<<dim3(NPTS / 16, CDIM / 16, BATCH), 32, 0, stream>>>(w2, b2, h2t, outp, stats3);
    k_gn3_residual <<<(BATCH * CDIM * NPTS) / 256, 256, 0, stream>>>(outp, g2, be2, stats3, x, (float*)d_out);
}